// SparseAttention_10436770529653
// MI455X (gfx1250) — compile-verified
//
#include <hip/hip_runtime.h>
#include <hip/hip_bf16.h>

// MI455X / gfx1250, wave32. fp32 end-to-end (memory-bound problem; fp32 WMMA
// 16x16x4 keeps reference precision). Fused conv->pool avoids materializing
// the 134MB full-res x2 tensor. Each conv block covers 64 out-channels so
// input2 is re-read only 4x (all L2-resident: 67MB << 192MB L2).

typedef __attribute__((ext_vector_type(2))) float v2f;
typedef __attribute__((ext_vector_type(8))) float v8f;

#define B_DIM   8
#define C_OUT   256
#define C_IN    128
#define WIMG    128
#define HIMG    128
#define ATT     32
#define NPOOL   (ATT * ATT)        // 1024 pooled pixels
#define PLANE   (WIMG * HIMG)      // 16384

// ---------------------------------------------------------------------------
// Kernel 1: pool input1 -> p1  (max + mean over exact 4x4 windows)
// One thread per pooled output; 4x global_load_b128 per thread.
// ---------------------------------------------------------------------------
__global__ __launch_bounds__(256) void pool_p1_kernel(
    const float* __restrict__ input1, float* __restrict__ p1)
{
    int idx = blockIdx.x * 256 + threadIdx.x;       // over B*C_OUT*NPOOL
    int p   = idx & (NPOOL - 1);
    int bc  = idx >> 10;
    int wo  = p >> 5, ho = p & 31;
    const float* src = input1 + (size_t)bc * PLANE + (wo * 4) * HIMG + (ho * 4);
    float mx = -3.402823466e38f, sm = 0.f;
#pragma unroll
    for (int dw = 0; dw < 4; ++dw) {
        float4 v = *(const float4*)(src + dw * HIMG);   // 16B-aligned row of 4
        mx = fmaxf(mx, fmaxf(fmaxf(v.x, v.y), fmaxf(v.z, v.w)));
        sm += (v.x + v.y) + (v.z + v.w);
    }
    p1[idx] = mx + sm * (1.0f / 16.0f);
}

// ---------------------------------------------------------------------------
// Kernel 2: fused 1x1-conv + pool -> p2, via V_WMMA_F32_16X16X4_F32.
// Block = 128 thr (4 waves): 64 pooled pixels x 64 out-channels.
// Wave tile: 16 pooled pixels (N), 4 co-sub-tiles of 16 (M), K = C_IN step 4.
// The 16 pool-window offsets (dw,dh) run as GEMM passes; max/sum accumulate
// elementwise per lane (C/D layout identical across passes). One set of B
// loads feeds 4 WMMAs (co-sub-tiles) -> 8 global loads per 16 WMMAs.
// Bias folded once at the end: max(x+b)+mean(x+b) = max(x)+mean(x)+2b.
// ---------------------------------------------------------------------------
__global__ __launch_bounds__(128) void conv_pool_p2_kernel(
    const float* __restrict__ input2, const float* __restrict__ conv_w,
    const float* __restrict__ conv_b, float* __restrict__ p2)
{
    __shared__ float wtile[64 * C_IN];              // 32 KB: W slice [64,128]
    const int lane     = threadIdx.x & 31;
    const int wave     = threadIdx.x >> 5;
    const int co_block = blockIdx.y * 64;
    const int batch    = blockIdx.z;

    // cooperative stage of conv_w[co_block:co_block+64, :] (vectorized)
    {
        const float4* wsrc = (const float4*)(conv_w + co_block * C_IN);
        float4*       wdst = (float4*)wtile;
        for (int j = threadIdx.x; j < (64 * C_IN) / 4; j += 128)
            wdst[j] = wsrc[j];
    }
    __syncthreads();

    const int n    = lane & 15;                     // N index (pooled pixel)
    const int half = lane >> 4;                     // K sub-split (16x16x4 layout)
    const int p    = (blockIdx.x * 4 + wave) * 16 + n;
    const int wo   = p >> 5, ho = p & 31;

    const float* src0 = input2 + (size_t)batch * C_IN * PLANE
                               + (size_t)(wo * 4) * HIMG + (ho * 4);

    v8f accmax[4], accsum[4];
#pragma unroll
    for (int t = 0; t < 4; ++t)
#pragma unroll
        for (int r = 0; r < 8; ++r) { accmax[t][r] = -3.402823466e38f; accsum[t][r] = 0.f; }

    for (int dw = 0; dw < 4; ++dw) {
        for (int dh = 0; dh < 4; ++dh) {
            const float* src = src0 + dw * HIMG + dh;
            v8f c0 = {}, c1 = {}, c2 = {}, c3 = {};
#pragma unroll 2
            for (int k0 = 0; k0 < C_IN; k0 += 4) {
                // B 4x16: N = lane&15 across lanes; K = k0 + 2*half {,+1}
                v2f bm;
                bm.x = src[(size_t)(k0 + 2 * half)     * PLANE];
                bm.y = src[(size_t)(k0 + 2 * half + 1) * PLANE];
                // A 16x4 per co-sub-tile: M = lane&15, same K split
                const int abase = n * C_IN + k0 + 2 * half;
                v2f a0 = *(const v2f*)&wtile[abase];
                v2f a1 = *(const v2f*)&wtile[abase + 16 * C_IN];
                v2f a2 = *(const v2f*)&wtile[abase + 32 * C_IN];
                v2f a3 = *(const v2f*)&wtile[abase + 48 * C_IN];
                c0 = __builtin_amdgcn_wmma_f32_16x16x4_f32(false, a0, false, bm, (short)0, c0, false, false);
                c1 = __builtin_amdgcn_wmma_f32_16x16x4_f32(false, a1, false, bm, (short)0, c1, false, false);
                c2 = __builtin_amdgcn_wmma_f32_16x16x4_f32(false, a2, false, bm, (short)0, c2, false, false);
                c3 = __builtin_amdgcn_wmma_f32_16x16x4_f32(false, a3, false, bm, (short)0, c3, false, false);
            }
            v8f cc[4] = {c0, c1, c2, c3};
#pragma unroll
            for (int t = 0; t < 4; ++t)
#pragma unroll
                for (int r = 0; r < 8; ++r) {
                    float v = cc[t][r];
                    accmax[t][r] = fmaxf(accmax[t][r], v);
                    accsum[t][r] += v;
                }
        }
    }

    // D layout: row M = r + 8*half, col N = lane&15
#pragma unroll
    for (int t = 0; t < 4; ++t)
#pragma unroll
        for (int r = 0; r < 8; ++r) {
            int co = co_block + t * 16 + r + 8 * half;
            float out = accmax[t][r] + accsum[t][r] * (1.0f / 16.0f) + 2.0f * conv_b[co];
            p2[((size_t)batch * C_OUT + co) * NPOOL + p] = out;
        }
}

// ---------------------------------------------------------------------------
// Kernel 3: per-position batch attention. One block (256 thr) per n.
// q=p1[:, :, n] [8,256], k=v=p2[:, :, n]; s = softmax(q k^T / sqrt(32));
// attn[b,c,n] = (s @ v)[b,c].
// ---------------------------------------------------------------------------
__global__ __launch_bounds__(256) void attention_kernel(
    const float* __restrict__ p1, const float* __restrict__ p2,
    float* __restrict__ attn)
{
    __shared__ float q[B_DIM][C_OUT];
    __shared__ float kv[B_DIM][C_OUT];
    __shared__ float s[B_DIM][B_DIM];
    const int n = blockIdx.x;
    const int t = threadIdx.x;

    for (int j = t; j < B_DIM * C_OUT; j += 256) {
        int bb = j >> 8, cc = j & 255;
        size_t g = ((size_t)bb * C_OUT + cc) * NPOOL + n;
        q[bb][cc]  = p1[g];
        kv[bb][cc] = p2[g];
    }
    __syncthreads();

    if (t < 64) {
        int bb = t >> 3, e = t & 7;
        float acc = 0.f;
#pragma unroll 8
        for (int cc = 0; cc < C_OUT; ++cc) acc += q[bb][cc] * kv[e][cc];
        s[bb][e] = acc * 0.17677669529663687f;      // 1/sqrt(256/8)
    }
    __syncthreads();

    if (t < 8) {
        float mx = s[t][0];
#pragma unroll
        for (int e = 1; e < 8; ++e) mx = fmaxf(mx, s[t][e]);
        float ex[8], sum = 0.f;
#pragma unroll
        for (int e = 0; e < 8; ++e) { ex[e] = __expf(s[t][e] - mx); sum += ex[e]; }
        float inv = 1.f / sum;
#pragma unroll
        for (int e = 0; e < 8; ++e) s[t][e] = ex[e] * inv;
    }
    __syncthreads();

    // thread t = channel c
#pragma unroll
    for (int bb = 0; bb < B_DIM; ++bb) {
        float acc = 0.f;
#pragma unroll
        for (int e = 0; e < 8; ++e) acc += s[bb][e] * kv[e][t];
        attn[((size_t)bb * C_OUT + t) * NPOOL + n] = acc;
    }
}

// ---------------------------------------------------------------------------
// Kernel 4: bilinear upsample 32->128 (align_corners=True) + sigmoid.
// One thread per output element; 4 gathers from a 4KB attn row (L0-resident).
// ---------------------------------------------------------------------------
__global__ __launch_bounds__(256) void upsample_sigmoid_kernel(
    const float* __restrict__ attn, float* __restrict__ out)
{
    int idx = blockIdx.x * 256 + threadIdx.x;
    int h  = idx & 127;
    int w  = (idx >> 7) & 127;
    int bc = idx >> 14;                              // b*256 + c
    const float sc = 31.0f / 127.0f;
    float pw = w * sc, ph = h * sc;
    int w0 = (int)pw, h0 = (int)ph;
    int w1 = min(w0 + 1, 31), h1 = min(h0 + 1, 31);
    float tw = pw - (float)w0, th = ph - (float)h0;
    const float* a = attn + (size_t)bc * NPOOL;
    float v00 = a[w0 * 32 + h0], v01 = a[w0 * 32 + h1];
    float v10 = a[w1 * 32 + h0], v11 = a[w1 * 32 + h1];
    float v0 = v00 + (v01 - v00) * th;
    float v1 = v10 + (v11 - v10) * th;
    float v  = v0 + (v1 - v0) * tw;
    out[idx] = 1.0f / (1.0f + __expf(-v));
}

// ---------------------------------------------------------------------------
extern "C" void kernel_launch(void* const* d_in, const int* in_sizes, int n_in,
                              void* d_out, int out_size, void* d_ws, size_t ws_size,
                              hipStream_t stream) {
    const float* input1 = (const float*)d_in[0];   // [8,256,128,128]
    const float* input2 = (const float*)d_in[1];   // [8,128,128,128]
    const float* conv_w = (const float*)d_in[2];   // [256,128]
    const float* conv_b = (const float*)d_in[3];   // [256]
    float* out = (float*)d_out;                    // [8,256,128,128]

    // workspace: p1 | p2 | attn, each B*C_OUT*NPOOL floats (8.4MB), total ~25MB
    float* p1   = (float*)d_ws;
    float* p2   = p1 + (size_t)B_DIM * C_OUT * NPOOL;
    float* attn = p2 + (size_t)B_DIM * C_OUT * NPOOL;

    pool_p1_kernel<<<(B_DIM * C_OUT * NPOOL) / 256, 256, 0, stream>>>(input1, p1);

    dim3 gconv(NPOOL / 64, C_OUT / 64, B_DIM);     // 16 x 4 x 8 blocks, 4 waves each
    conv_pool_p2_kernel<<<gconv, 128, 0, stream>>>(input2, conv_w, conv_b, p2);

    attention_kernel<<<NPOOL, 256, 0, stream>>>(p1, p2, attn);

    upsample_sigmoid_kernel<<<(B_DIM * C_OUT * PLANE) / 256, 256, 0, stream>>>(attn, out);
}